// DynamicFrameSelectionModule_16252156248185
// MI455X (gfx1250) — compile-verified
//
#include <hip/hip_runtime.h>
#include <math.h>

#define HW    50176          // 224*224
#define HW4   (HW/4)         // 12544 float4 positions
#define TDIM  64
#define BDIM  8
#define CIN   3
#define COUT  4

typedef __attribute__((ext_vector_type(2))) float v2f;
typedef __attribute__((ext_vector_type(8))) float v8f;

// ---------------------------------------------------------------------------
// Pass 1: s[b,t] = mean over (o,h,w) of relu(conv_w[o,:]·x[b,:,t,h,w]+conv_b[o])
// One block per (b,t) slice; streams 3*HW floats, reduces to one scalar.
// ---------------------------------------------------------------------------
__global__ __launch_bounds__(256) void k_reduce(const float* __restrict__ x,
                                                const float* __restrict__ cw,
                                                const float* __restrict__ cb,
                                                float* __restrict__ s_out) {
  const int bt = blockIdx.x;
  const int b = bt >> 6, t = bt & 63;

  const float w00=cw[0],  w01=cw[1],  w02=cw[2];
  const float w10=cw[3],  w11=cw[4],  w12=cw[5];
  const float w20=cw[6],  w21=cw[7],  w22=cw[8];
  const float w30=cw[9],  w31=cw[10], w32=cw[11];
  const float c0=cb[0], c1=cb[1], c2=cb[2], c3=cb[3];

  const float4* __restrict__ x0 = (const float4*)(x + ((size_t)(b*CIN+0)*TDIM + t) * HW);
  const float4* __restrict__ x1 = (const float4*)(x + ((size_t)(b*CIN+1)*TDIM + t) * HW);
  const float4* __restrict__ x2 = (const float4*)(x + ((size_t)(b*CIN+2)*TDIM + t) * HW);

  float acc = 0.f;
  for (int p = threadIdx.x; p < HW4; p += 256) {
    __builtin_prefetch(&x0[p + 256], 0, 0);
    __builtin_prefetch(&x1[p + 256], 0, 0);
    __builtin_prefetch(&x2[p + 256], 0, 0);
    float4 a = x0[p], bb = x1[p], d = x2[p];
    const float ax[4] = {a.x, a.y, a.z, a.w};
    const float bx[4] = {bb.x, bb.y, bb.z, bb.w};
    const float dx[4] = {d.x, d.y, d.z, d.w};
#pragma unroll
    for (int j = 0; j < 4; ++j) {
      const float u0 = ax[j], u1 = bx[j], u2 = dx[j];
      acc += fmaxf(0.f, fmaf(w00,u0, fmaf(w01,u1, fmaf(w02,u2, c0))));
      acc += fmaxf(0.f, fmaf(w10,u0, fmaf(w11,u1, fmaf(w12,u2, c1))));
      acc += fmaxf(0.f, fmaf(w20,u0, fmaf(w21,u1, fmaf(w22,u2, c2))));
      acc += fmaxf(0.f, fmaf(w30,u0, fmaf(w31,u1, fmaf(w32,u2, c3))));
    }
  }

  // wave32 reduction, then cross-wave via LDS
  __shared__ float red[8];
  float v = acc;
#pragma unroll
  for (int off = 16; off; off >>= 1) v += __shfl_down(v, off);
  const int lane = threadIdx.x & 31, wid = threadIdx.x >> 5;
  if (lane == 0) red[wid] = v;
  __syncthreads();
  if (threadIdx.x == 0) {
    float tot = 0.f;
#pragma unroll
    for (int i = 0; i < 8; ++i) tot += red[i];
    s_out[bt] = tot * (1.f / (float)(COUT * HW));
  }
}

// ---------------------------------------------------------------------------
// Pass 2: SE MLP on the matrix pipe (v_wmma_f32_16x16x4_f32) + dual top-4.
// Single wave32 block. S padded to 16x64 (rows 8..15 zero).
//   H(16x32)      = relu(S(16x64) @ W1(64x32) + b1)
//   scores(8x64)  = sigmoid(H(16x32) @ W2(32x64) + b2)   (rows >= 8 discarded)
// ---------------------------------------------------------------------------
__global__ __launch_bounds__(32) void k_mlp_topk(const float* __restrict__ s_in,
                                                 const float* __restrict__ w1,
                                                 const float* __restrict__ b1,
                                                 const float* __restrict__ w2,
                                                 const float* __restrict__ b2,
                                                 int* __restrict__ idx_out) {
  __shared__ float sA[16 * 64];
  __shared__ float sH[16 * 32];
  __shared__ float sS[8 * 64];

  const int lane = threadIdx.x;
  for (int i = lane; i < 16 * 64; i += 32) {
    const int m = i >> 6, k = i & 63;
    sA[i] = (m < 8) ? s_in[m * 64 + k] : 0.f;
  }
  __syncthreads();

  // fragment coordinates (ISA 7.12.2: f32 16x4 A layout, 16x16 f32 C/D layout)
  const int M    = lane & 15;          // A row
  const int koff = (lane >> 4) << 1;   // lanes 16..31 hold K+2,K+3
  const int moff = (lane >> 4) << 3;   // C/D: lanes 16..31 hold M+8
  const int N    = lane & 15;          // B/C column within tile

  // GEMM1: 2 N-tiles, K=64 -> 16 WMMAs each
  for (int n0 = 0; n0 < 32; n0 += 16) {
    v8f c = {};
    for (int k0 = 0; k0 < 64; k0 += 4) {
      v2f a, bf;
      a.x  = sA[M * 64 + k0 + koff];
      a.y  = sA[M * 64 + k0 + koff + 1];
      bf.x = w1[(k0 + koff)     * 32 + n0 + N];
      bf.y = w1[(k0 + koff + 1) * 32 + n0 + N];
      c = __builtin_amdgcn_wmma_f32_16x16x4_f32(false, a, false, bf,
                                                (short)0, c, false, false);
    }
#pragma unroll
    for (int i = 0; i < 8; ++i) {
      const int mm = i + moff, nn = n0 + N;
      sH[mm * 32 + nn] = fmaxf(c[i] + b1[nn], 0.f);
    }
  }
  __syncthreads();

  // GEMM2: 4 N-tiles, K=32 -> 8 WMMAs each
  for (int n0 = 0; n0 < 64; n0 += 16) {
    v8f c = {};
    for (int k0 = 0; k0 < 32; k0 += 4) {
      v2f a, bf;
      a.x  = sH[M * 32 + k0 + koff];
      a.y  = sH[M * 32 + k0 + koff + 1];
      bf.x = w2[(k0 + koff)     * 64 + n0 + N];
      bf.y = w2[(k0 + koff + 1) * 64 + n0 + N];
      c = __builtin_amdgcn_wmma_f32_16x16x4_f32(false, a, false, bf,
                                                (short)0, c, false, false);
    }
#pragma unroll
    for (int i = 0; i < 8; ++i) {
      const int mm = i + moff, nn = n0 + N;
      if (mm < 8) {
        const float v = c[i] + b2[nn];
        sS[mm * 64 + nn] = 1.f / (1.f + __expf(-v));
      }
    }
  }
  __syncthreads();

  // Dual top-4 per batch; strict compares give jax.lax.top_k stable tie order.
  if (lane < 8) {
    float tv0=-1e30f, tv1=-1e30f, tv2=-1e30f, tv3=-1e30f;
    int   ti0=0, ti1=0, ti2=0, ti3=0;
    float bv0= 1e30f, bv1= 1e30f, bv2= 1e30f, bv3= 1e30f;
    int   bi0=0, bi1=0, bi2=0, bi3=0;
    for (int t = 0; t < TDIM; ++t) {
      const float v = sS[lane * 64 + t];
      if      (v > tv0) { tv3=tv2; ti3=ti2; tv2=tv1; ti2=ti1; tv1=tv0; ti1=ti0; tv0=v; ti0=t; }
      else if (v > tv1) { tv3=tv2; ti3=ti2; tv2=tv1; ti2=ti1; tv1=v;  ti1=t; }
      else if (v > tv2) { tv3=tv2; ti3=ti2; tv2=v;  ti2=t; }
      else if (v > tv3) { tv3=v;  ti3=t; }
      if      (v < bv0) { bv3=bv2; bi3=bi2; bv2=bv1; bi2=bi1; bv1=bv0; bi1=bi0; bv0=v; bi0=t; }
      else if (v < bv1) { bv3=bv2; bi3=bi2; bv2=bv1; bi2=bi1; bv1=v;  bi1=t; }
      else if (v < bv2) { bv3=bv2; bi3=bi2; bv2=v;  bi2=t; }
      else if (v < bv3) { bv3=v;  bi3=t; }
    }
    int* o = idx_out + lane * 8;
    o[0]=ti0; o[1]=ti1; o[2]=ti2; o[3]=ti3;   // top, descending
    o[4]=bi0; o[5]=bi1; o[6]=bi2; o[7]=bi3;   // bottom, ascending
  }
}

// ---------------------------------------------------------------------------
// Pass 3: recompute conv+ReLU only for the 8 selected frames per batch and
// scatter to out[b, o, j, h, w]. 512 blocks: (b,j) x 8 chunks of HW.
// ---------------------------------------------------------------------------
__device__ __forceinline__ float4 conv_relu4(float4 a, float4 b, float4 d,
                                             float w0, float w1, float w2, float bb) {
  float4 r;
  r.x = fmaxf(0.f, fmaf(w0,a.x, fmaf(w1,b.x, fmaf(w2,d.x, bb))));
  r.y = fmaxf(0.f, fmaf(w0,a.y, fmaf(w1,b.y, fmaf(w2,d.y, bb))));
  r.z = fmaxf(0.f, fmaf(w0,a.z, fmaf(w1,b.z, fmaf(w2,d.z, bb))));
  r.w = fmaxf(0.f, fmaf(w0,a.w, fmaf(w1,b.w, fmaf(w2,d.w, bb))));
  return r;
}

__global__ __launch_bounds__(256) void k_gather(const float* __restrict__ x,
                                                const float* __restrict__ cw,
                                                const float* __restrict__ cb,
                                                const int* __restrict__ idx,
                                                float* __restrict__ out) {
  const int chunk = blockIdx.x & 7;
  const int fj    = blockIdx.x >> 3;   // 0..63
  const int b     = fj >> 3, j = fj & 7;
  const int t     = idx[b * 8 + j];

  const float w00=cw[0],  w01=cw[1],  w02=cw[2];
  const float w10=cw[3],  w11=cw[4],  w12=cw[5];
  const float w20=cw[6],  w21=cw[7],  w22=cw[8];
  const float w30=cw[9],  w31=cw[10], w32=cw[11];
  const float c0=cb[0], c1=cb[1], c2=cb[2], c3=cb[3];

  const float4* __restrict__ x0 = (const float4*)(x + ((size_t)(b*CIN+0)*TDIM + t) * HW);
  const float4* __restrict__ x1 = (const float4*)(x + ((size_t)(b*CIN+1)*TDIM + t) * HW);
  const float4* __restrict__ x2 = (const float4*)(x + ((size_t)(b*CIN+2)*TDIM + t) * HW);
  float4* __restrict__ o0 = (float4*)(out + ((size_t)((b*COUT+0)*8 + j)) * HW);
  float4* __restrict__ o1 = (float4*)(out + ((size_t)((b*COUT+1)*8 + j)) * HW);
  float4* __restrict__ o2 = (float4*)(out + ((size_t)((b*COUT+2)*8 + j)) * HW);
  float4* __restrict__ o3 = (float4*)(out + ((size_t)((b*COUT+3)*8 + j)) * HW);

  const int p0 = chunk * (HW4 / 8), p1 = p0 + (HW4 / 8);
  for (int p = p0 + threadIdx.x; p < p1; p += 256) {
    float4 a = x0[p], bb = x1[p], d = x2[p];
    o0[p] = conv_relu4(a, bb, d, w00, w01, w02, c0);
    o1[p] = conv_relu4(a, bb, d, w10, w11, w12, c1);
    o2[p] = conv_relu4(a, bb, d, w20, w21, w22, c2);
    o3[p] = conv_relu4(a, bb, d, w30, w31, w32, c3);
  }
}

// ---------------------------------------------------------------------------
extern "C" void kernel_launch(void* const* d_in, const int* in_sizes, int n_in,
                              void* d_out, int out_size, void* d_ws, size_t ws_size,
                              hipStream_t stream) {
  const float* x      = (const float*)d_in[0];
  const float* conv_w = (const float*)d_in[1];
  const float* conv_b = (const float*)d_in[2];
  const float* w1     = (const float*)d_in[3];
  const float* b1     = (const float*)d_in[4];
  const float* w2     = (const float*)d_in[5];
  const float* b2     = (const float*)d_in[6];
  float* out = (float*)d_out;

  float* s_ws  = (float*)d_ws;                                   // 512 f32
  int*   i_ws  = (int*)((char*)d_ws + BDIM * TDIM * sizeof(float)); // 64 i32

  k_reduce  <<<BDIM * TDIM, 256, 0, stream>>>(x, conv_w, conv_b, s_ws);
  k_mlp_topk<<<1,           32,  0, stream>>>(s_ws, w1, b1, w2, b2, i_ws);
  k_gather  <<<512,         256, 0, stream>>>(x, conv_w, conv_b, i_ws, out);
}